// WSeq_attention_16681652978160
// MI455X (gfx1250) — compile-verified
//
#include <hip/hip_runtime.h>
#include <math.h>

// WSeq attention: cosine-sim softmax pooling over T=50 utterances + self.
//   query:      [B, H]      f32   (d_in[0])
//   utterances: [T, B, H]   f32   (d_in[1])
//   out:        [B, H]      f32
// One workgroup per batch element b. Single HBM pass over utterances:
// stage V = [utt_b (50 rows); query (row 50); zeros (rows 51..63)] in LDS
// via GLOBAL_LOAD_ASYNC_TO_LDS_B128 (ASYNCcnt-tracked, no VGPR round-trip),
// compute Gram tiles with V_WMMA_F32_16X16X4_F32 (dots = column 50,
// norms^2 = diagonal), softmax in one wave, weighted sum out of LDS.

typedef float v2f __attribute__((ext_vector_type(2)));
typedef float v8f __attribute__((ext_vector_type(8)));
typedef int   v4i __attribute__((vector_size(16)));      // matches builtin's V4i

typedef __attribute__((address_space(1))) v4i g_v4i;     // global int4
typedef __attribute__((address_space(3))) v4i l_v4i;     // LDS int4

#define T_UT   50
#define HDIM   512
#define LDW    516      // padded LDS row stride (floats); 516 % 64 == 4 -> bank-conflict-free
#define NROWS  64
#define BLOCK  512      // 16 wave32s

#if defined(__has_builtin)
#if __has_builtin(__builtin_amdgcn_global_load_async_to_lds_b128)
#define USE_ASYNC_LDS 1
#endif
#endif

__device__ __forceinline__ void async_cp16(const float* gsrc, float* ldst) {
#if defined(USE_ASYNC_LDS)
  __builtin_amdgcn_global_load_async_to_lds_b128(
      (g_v4i*)gsrc, (l_v4i*)ldst, /*offset=*/0, /*cpol=*/0);
#else
  *reinterpret_cast<float4*>(ldst) = *reinterpret_cast<const float4*>(gsrc);
#endif
}

__device__ __forceinline__ void async_fence() {
#if defined(USE_ASYNC_LDS)
#if defined(__has_builtin)
#if __has_builtin(__builtin_amdgcn_s_wait_asynccnt)
  __builtin_amdgcn_s_wait_asynccnt(0);
#else
  asm volatile("s_wait_asynccnt 0x0" ::: "memory");
#endif
#endif
#endif
}

__global__ __launch_bounds__(BLOCK)
void wseq_attn_kernel(const float* __restrict__ q,
                      const float* __restrict__ utt,
                      float* __restrict__ out, int B) {
  __shared__ float V[NROWS * LDW];   // 64 x 516 f32 = 129 KB
  __shared__ float dots[64];
  __shared__ float n2[64];
  __shared__ float w[64];

  const int b    = blockIdx.x;
  const int tid  = threadIdx.x;
  const int lane = tid & 31;
  const int wave = tid >> 5;

  // ---- Phase A: stage rows 0..49 (utterances) + row 50 (query) into LDS ----
  // Async HBM->LDS copy, one b128 per lane per step; tracked by ASYNCcnt.
  for (int i = tid; i < 51 * (HDIM / 4); i += BLOCK) {
    int row = i >> 7;                 // 128 float4 per row
    int col = (i & 127) << 2;
    const float* src = (row < T_UT)
        ? (utt + ((size_t)row * B + b) * HDIM + col)
        : (q + (size_t)b * HDIM + col);
    async_cp16(src, &V[row * LDW + col]);
  }
  // zero-pad rows 51..63 so the tail WMMA tiles are well-defined
  for (int i = tid + 51 * 128; i < 64 * 128; i += BLOCK) {
    int row = i >> 7;
    int col = (i & 127) << 2;
    *reinterpret_cast<float4*>(&V[row * LDW + col]) = make_float4(0.f, 0.f, 0.f, 0.f);
  }
  async_fence();
  __syncthreads();

  // ---- Phase B: Gram tiles G = V * V^T via V_WMMA_F32_16X16X4_F32 ----
  // waves 0..3: tiles (m,3)  -> dots column n=50 (+ diag for tile (3,3))
  // waves 4..6: tiles (m,m)  -> diagonal (norms^2) for rows 0..47
  if (wave < 7) {
    const int mt = (wave < 4) ? wave : (wave - 4);
    const int nt = (wave < 4) ? 3 : mt;
    const int li = lane & 15;        // A: M = li ; B: N = li
    const int hf = lane >> 4;        // K pair select: K = 2*hf + vgpr
    const float* pa = &V[(mt * 16 + li) * LDW + 2 * hf];
    const float* pb = &V[(nt * 16 + li) * LDW + 2 * hf];
    v8f c = {0.f, 0.f, 0.f, 0.f, 0.f, 0.f, 0.f, 0.f};
#pragma unroll 8
    for (int kc = 0; kc < HDIM / 4; ++kc) {
      v2f a  = *reinterpret_cast<const v2f*>(pa + kc * 4);
      v2f bb = *reinterpret_cast<const v2f*>(pb + kc * 4);
      // 8 args: (neg_a, A, neg_b, B, c_mod, C, reuse_a, reuse_b)
      c = __builtin_amdgcn_wmma_f32_16x16x4_f32(false, a, false, bb,
                                                (short)0, c, false, false);
    }
    // dots: column n=50 lives at n_local=2 of tile column 3 (lanes 2 and 18)
    if (nt == 3 && li == 2) {
      int mbase = mt * 16 + 8 * hf;  // C vgpr r holds M = r + 8*hf
#pragma unroll
      for (int r = 0; r < 8; ++r) dots[mbase + r] = c[r];
    }
    // diagonal (norms^2): element (d,d) is held where N==M
    if (mt == nt) {
#pragma unroll
      for (int r = 0; r < 8; ++r) {
        int d = r + 8 * hf;          // M for this C vgpr
        if (d == li) n2[nt * 16 + li] = c[r];
      }
    }
  }
  __syncthreads();

  // ---- Phase C: cosine scores + softmax over T+1 = 51 entries (wave 0) ----
  if (wave == 0) {
    float rq = rsqrtf(n2[50]);                          // 1/||q||
    float sA = dots[lane] * rq * rsqrtf(n2[lane]);      // t = lane (0..31 < 50)
    int   t2 = lane + 32;
    float sB;
    if (t2 < T_UT)       sB = dots[t2] * rq * rsqrtf(n2[t2]);
    else if (t2 == T_UT) sB = 1.0f;                     // appended self-score
    else                 sB = -3.0e38f;
    float m = fmaxf(sA, sB);
#pragma unroll
    for (int off = 16; off > 0; off >>= 1) m = fmaxf(m, __shfl_xor(m, off, 32));
    float eA = __expf(sA - m);
    float eB = (t2 <= T_UT) ? __expf(sB - m) : 0.0f;
    float s = eA + eB;
#pragma unroll
    for (int off = 16; off > 0; off >>= 1) s += __shfl_xor(s, off, 32);
    float inv = 1.0f / s;
    w[lane] = eA * inv;
    if (t2 <= T_UT) w[t2] = eB * inv;
  }
  __syncthreads();

  // ---- Phase D: out[b,h] = sum_{t=0..50} w[t] * V[t,h]  (h = tid) ----
  {
    int h = tid;
    float acc = 0.f;
#pragma unroll 3
    for (int t = 0; t <= T_UT; ++t) acc = fmaf(w[t], V[t * LDW + h], acc);
    out[(size_t)b * HDIM + h] = acc;
  }
}

extern "C" void kernel_launch(void* const* d_in, const int* in_sizes, int n_in,
                              void* d_out, int out_size, void* d_ws, size_t ws_size,
                              hipStream_t stream) {
  (void)n_in; (void)out_size; (void)d_ws; (void)ws_size;
  const float* q   = (const float*)d_in[0];   // [B, H]
  const float* utt = (const float*)d_in[1];   // [T, B, H]
  float* out = (float*)d_out;                 // [B, H]
  int B = in_sizes[0] / HDIM;                 // 4096
  wseq_attn_kernel<<<B, BLOCK, 0, stream>>>(q, utt, out, B);
}